// CoulombLR_DSF_NB_47991964566176
// MI455X (gfx1250) — compile-verified
//
#include <hip/hip_runtime.h>
#include <stdint.h>

// ---- module constants from the reference ----
#define ALPHA_F      0.2f
#define RC_SHORT_F   4.6f
#define KE_F         7.1998226f
#define INV_SQRT_PI  0.5641895835477563f   // 1/sqrt(pi)
#define MAX_NB       64
#define WAVES_PER_BLOCK 8

// Hardware transcendental approximations (single VALU trans op each).
__device__ __forceinline__ float frcp(float x)  { return __builtin_amdgcn_rcpf(x);  }
__device__ __forceinline__ float frsq(float x)  { return __builtin_amdgcn_rsqf(x);  }
__device__ __forceinline__ float fsqrt(float x) { return __builtin_amdgcn_sqrtf(x); }

// erfc(x) for x >= 0: Abramowitz & Stegun 7.1.26, |err| < 1.5e-7.
// Cost: 1 v_rcp + 1 v_exp + ~7 FMA/MUL.
__device__ __forceinline__ float erfc_fast(float x) {
    const float p  = 0.3275911f;
    const float a1 = 0.254829592f, a2 = -0.284496736f, a3 = 1.421413741f,
                a4 = -1.453152027f, a5 = 1.061405429f;
    float t    = frcp(fmaf(p, x, 1.0f));
    float poly = t * fmaf(t, fmaf(t, fmaf(t, fmaf(t, a5, a4), a3), a2), a1);
    return poly * __expf(-x * x);            // __expf -> v_exp_f32
}

// Low 32 bits of a flat shared-aperture address are the LDS byte offset
// (ISA 10.2); that is the VDST operand for
// global_load_async_to_lds_*.
__device__ __forceinline__ uint32_t lds_off(const void* p) {
    return (uint32_t)(uintptr_t)p;
}

// Per-wave staging buffer for one atom's streaming rows (640 B).
struct WaveBuf {
    int     idxJ[MAX_NB];   // 256 B
    int     idxC[MAX_NB];   // 256 B
    uint8_t mS[MAX_NB];     //  64 B
    uint8_t mC[MAX_NB];     //  64 B
};

// Issue 6 async DMA ops per lane staging atom `i`'s rows into `b`.
// b64 x2 cover the two 256B index rows (32 lanes x 8B); b8 x4 cover the
// two 64B mask rows (instruction offset applies to BOTH global and LDS
// addresses, so offset:32 reuses the same address operands).
__device__ __forceinline__ void issue_async_batch(const int*     idx_j,
                                                  const int*     idx_jc,
                                                  const uint8_t* mask_s,
                                                  const uint8_t* mask_c,
                                                  long long nb, int lane,
                                                  WaveBuf* b)
{
    const int*     gJ = idx_j  + nb + 2 * lane;
    const int*     gC = idx_jc + nb + 2 * lane;
    const uint8_t* gS = mask_s + nb + lane;
    const uint8_t* gM = mask_c + nb + lane;
    uint32_t lJ = lds_off(&b->idxJ[2 * lane]);
    uint32_t lC = lds_off(&b->idxC[2 * lane]);
    uint32_t lS = lds_off(&b->mS[lane]);
    uint32_t lM = lds_off(&b->mC[lane]);
    asm volatile("global_load_async_to_lds_b64 %0, %1, off"
                 :: "v"(lJ), "v"(gJ) : "memory");
    asm volatile("global_load_async_to_lds_b64 %0, %1, off"
                 :: "v"(lC), "v"(gC) : "memory");
    asm volatile("global_load_async_to_lds_b8 %0, %1, off"
                 :: "v"(lS), "v"(gS) : "memory");
    asm volatile("global_load_async_to_lds_b8 %0, %1, off offset:32"
                 :: "v"(lS), "v"(gS) : "memory");
    asm volatile("global_load_async_to_lds_b8 %0, %1, off"
                 :: "v"(lM), "v"(gM) : "memory");
    asm volatile("global_load_async_to_lds_b8 %0, %1, off offset:32"
                 :: "v"(lM), "v"(gM) : "memory");
}

// One wave32 per atom; grid-stride loop with double-buffered async staging.
__global__ __launch_bounds__(WAVES_PER_BLOCK * 32) void
dsf_coulomb_kernel(const float*   __restrict__ coord,     // [N,3]
                   const float*   __restrict__ charges,   // [N]
                   const int*     __restrict__ idx_j,     // [N,64]
                   const int*     __restrict__ idx_jc,    // [N,64]
                   const uint8_t* __restrict__ mask_s,    // [N,64] bool
                   const uint8_t* __restrict__ mask_c,    // [N,64] bool
                   const float*   __restrict__ rc_ptr,    // [1]
                   float*         __restrict__ out,       // [N]
                   int n_atoms)
{
    __shared__ WaveBuf sbuf[WAVES_PER_BLOCK][2];   // 10 KB / block

    const int lane       = threadIdx.x & 31;
    const int wave       = threadIdx.x >> 5;
    const int waveId     = blockIdx.x * WAVES_PER_BLOCK + wave;
    const int waveStride = gridDim.x * WAVES_PER_BLOCK;

    // Wave-uniform DSF shift constants.
    const float Rc    = rc_ptr[0];
    const float invRc = frcp(Rc);
    const float aRc   = ALPHA_F * Rc;
    const float c2    = erfc_fast(aRc) * invRc;
    const float c34   = fmaf(2.0f * ALPHA_F * __expf(-aRc * aRc), INV_SQRT_PI * invRc,
                             c2 * invRc);          // c3 + c4
    const float rc2s  = RC_SHORT_F * RC_SHORT_F;

    int buf = 0;
    if (waveId < n_atoms)
        issue_async_batch(idx_j, idx_jc, mask_s, mask_c,
                          (long long)waveId * MAX_NB, lane, &sbuf[wave][0]);

    for (int i = waveId; i < n_atoms; i += waveStride) {
        const int inext = i + waveStride;
        WaveBuf* cur = &sbuf[wave][buf];

        if (inext < n_atoms) {
            // launch next atom's DMA, then wait for current batch only:
            // 6 ops outstanding from the new batch are allowed.
            issue_async_batch(idx_j, idx_jc, mask_s, mask_c,
                              (long long)inext * MAX_NB, lane,
                              &sbuf[wave][buf ^ 1]);
            asm volatile("s_wait_asynccnt 0x6" ::: "memory");
        } else {
            asm volatile("s_wait_asynccnt 0x0" ::: "memory");
        }

        const float xi = coord[3 * i + 0];
        const float yi = coord[3 * i + 1];
        const float zi = coord[3 * i + 2];

        float acc = 0.0f;   // (dsf_term - short_term); KE*q_i factored out

        #pragma unroll
        for (int s = 0; s < 2; ++s) {
            const int  m  = lane + 32 * s;
            const int  j  = cur->idxJ[m];
            const int  jc = cur->idxC[m];
            const bool mC = cur->mC[m] != 0;   // coul pad mask (also masks short d, per ref)
            const bool mS = cur->mS[m] != 0;   // short pad mask (zeroes q_j only)

            // ---------- short-range exp-cutoff pass ----------
            {
                float dx = coord[3 * j + 0] - xi;
                float dy = coord[3 * j + 1] - yi;
                float dz = coord[3 * j + 2] - zi;
                float d2 = fmaf(dx, dx, fmaf(dy, dy, dz * dz));
                d2 = mC ? 1.0f : d2;                       // padded slots -> d = 1
                float invd = frsq(d2);                     // 1/d
                float fc = (d2 < rc2s) ? __expf(-d2 * frcp(rc2s - d2)) : 0.0f;
                float qj = mS ? 0.0f : charges[j];
                acc = fmaf(-fc * invd, qj, acc);
            }

            // ---------- DSF long-range pass ----------
            {
                float dx = coord[3 * jc + 0] - xi;
                float dy = coord[3 * jc + 1] - yi;
                float dz = coord[3 * jc + 2] - zi;
                float d2 = fmaf(dx, dx, fmaf(dy, dy, dz * dz));
                d2 = mC ? 1.0f : d2;
                float d    = fsqrt(d2);                    // d=0 -> inf (matches ref)
                float invd = frcp(d);
                float qj   = (mC || (d > Rc)) ? 0.0f : charges[jc];
                float c1   = erfc_fast(ALPHA_F * d) * invd;
                float term = fmaf(d - Rc, c34, c1 - c2);
                acc = fmaf(qj, term, acc);
            }
        }

        // wave32 butterfly reduction
        #pragma unroll
        for (int off = 16; off > 0; off >>= 1)
            acc += __shfl_xor(acc, off, 32);

        if (lane == 0)
            out[i] = KE_F * charges[i] * acc;

        buf ^= 1;
    }
}

extern "C" void kernel_launch(void* const* d_in, const int* in_sizes, int n_in,
                              void* d_out, int out_size, void* d_ws, size_t ws_size,
                              hipStream_t stream)
{
    // setup_inputs() order:
    const float*   coord   = (const float*)  d_in[0];  // [N,3] f32
    const float*   charges = (const float*)  d_in[1];  // [N]   f32
    const int*     idx_j   = (const int*)    d_in[2];  // [N,64] i32
    const int*     idx_jc  = (const int*)    d_in[3];  // [N,64] i32
    const uint8_t* mask_s  = (const uint8_t*)d_in[4];  // [N,64] bool (1B)
    const uint8_t* mask_c  = (const uint8_t*)d_in[5];  // [N,64] bool (1B)
    const float*   rc      = (const float*)  d_in[6];  // [1]   f32
    float*         out     = (float*)d_out;

    const int n = in_sizes[1];                 // N_ATOMS

    const int threads = WAVES_PER_BLOCK * 32;  // 8 wave32s per block
    int blocks = 4096;                         // grid-stride: ~6 atoms/wave -> double buffering pays off
    const int needed = (n + WAVES_PER_BLOCK - 1) / WAVES_PER_BLOCK;
    if (blocks > needed) blocks = needed;

    dsf_coulomb_kernel<<<blocks, threads, 0, stream>>>(
        coord, charges, idx_j, idx_jc, mask_s, mask_c, rc, out, n);
}